// PerPixelConv_76373108457968
// MI455X (gfx1250) — compile-verified
//
#include <hip/hip_runtime.h>
#include <stdint.h>

// Per-pixel 5x5 convolution: out[b,h,w] = sum_{kw,kh} kernel[b,kw*5+kh,h,w] * imgpad[b,h+kh,w+kw]
// Memory-bound streaming kernel for MI455X (roofline: 0.46 FLOP/byte, ~19us HBM floor):
//  - image tile staged into LDS via gfx1250 async global->LDS DMA (ASYNCcnt)
//  - 419MB weight stream via non-temporal b128 loads (protect image L2 residency)
//  - NT b128 output stores

typedef float v4f __attribute__((ext_vector_type(4)));
typedef int   v4i __attribute__((__vector_size__(16)));   // matches builtin prototype

#define KK 5
#define NB 16
#define NH 512
#define NW 512
#define TILE_H 2
#define TROWS (TILE_H + 4)      // 6 image rows (2 rows + 2-halo top/bottom)
#define TSTRIDE (NW + 8)        // 520: interior starts at LDS col 4 -> 16B aligned
#define NTHREADS 256

__device__ __forceinline__ void async_f4_to_lds(const float* __restrict__ g, float* lds) {
#if __has_builtin(__builtin_amdgcn_global_load_async_to_lds_b128)
  __builtin_amdgcn_global_load_async_to_lds_b128(
      (__attribute__((address_space(1))) v4i*)(uintptr_t)g,
      (__attribute__((address_space(3))) v4i*)lds,
      0, 0);
#else
  asm volatile("global_load_async_to_lds_b128 %0, %1, off"
               :
               : "v"((__attribute__((address_space(3))) void*)lds),
                 "v"(g)
               : "memory");
#endif
}

__device__ __forceinline__ void wait_asynccnt0() {
#if __has_builtin(__builtin_amdgcn_s_wait_asynccnt)
  __builtin_amdgcn_s_wait_asynccnt(0);
#else
  asm volatile("s_wait_asynccnt 0" ::: "memory");
#endif
}

__global__ __launch_bounds__(NTHREADS)
void ppconv_kernel(const float* __restrict__ kern,
                   const float* __restrict__ img,
                   float* __restrict__ out) {
  __shared__ __align__(16) float tile[TROWS * TSTRIDE];

  const int b   = blockIdx.y;
  const int h0  = blockIdx.x * TILE_H;
  const int tid = threadIdx.x;

  // ---- 1) zero the LDS tile (halo columns / out-of-range rows stay zero) ----
  {
    v4f z = {0.f, 0.f, 0.f, 0.f};
#pragma unroll
    for (int i = 0; i < 4; ++i) {
      int idx4 = (tid + i * NTHREADS) * 4;
      if (idx4 < TROWS * TSTRIDE) *(v4f*)&tile[idx4] = z;
    }
  }
  __syncthreads();

  // ---- 2) async DMA in-bounds image rows into the tile interior ----
  // 6 rows x 128 float4 columns = 768 transfers, 3 per thread
#pragma unroll
  for (int i = 0; i < 3; ++i) {
    int t    = tid + i * NTHREADS;   // 0..767
    int row  = t >> 7;               // 0..5
    int cv   = (t & 127) << 2;       // 0,4,...,508
    int grow = h0 - 2 + row;
    if (grow >= 0 && grow < NH) {
      const float* src = img + ((size_t)b * NH + grow) * NW + cv;
      float* dst = &tile[row * TSTRIDE + 4 + cv];   // 16B-aligned LDS dest
      async_f4_to_lds(src, dst);
    }
  }
  wait_asynccnt0();
  __syncthreads();

  // ---- 3) compute: 4 output pixels per thread ----
  const int r = tid >> 7;            // row within tile: 0..1
  const int w = (tid & 127) << 2;    // col: 0..508
  const int h = h0 + r;

  v4f acc = {0.f, 0.f, 0.f, 0.f};
  const size_t chanStride = (size_t)NH * NW;
  const size_t kbase = ((size_t)b * 25) * chanStride + (size_t)h * NW + w;

#pragma unroll
  for (int kh = 0; kh < KK; ++kh) {
    // 8 consecutive taps cover all 5 kw shifts for 4 output pixels
    float t8[8];
    const float* trow = &tile[(r + kh) * TSTRIDE + 2 + w];
#pragma unroll
    for (int i = 0; i < 8; ++i) t8[i] = trow[i];
#pragma unroll
    for (int kw = 0; kw < KK; ++kw) {
      const int c = kw * KK + kh;
      v4f wv = __builtin_nontemporal_load(
          (const v4f*)(kern + kbase + (size_t)c * chanStride));
      acc.x += wv.x * t8[kw + 0];
      acc.y += wv.y * t8[kw + 1];
      acc.z += wv.z * t8[kw + 2];
      acc.w += wv.w * t8[kw + 3];
    }
  }

  float* op = out + ((size_t)b * NH + h) * NW + w;
  __builtin_nontemporal_store(acc, (v4f*)op);
}

extern "C" void kernel_launch(void* const* d_in, const int* in_sizes, int n_in,
                              void* d_out, int out_size, void* d_ws, size_t ws_size,
                              hipStream_t stream) {
  (void)in_sizes; (void)n_in; (void)out_size; (void)d_ws; (void)ws_size;
  const float* kern = (const float*)d_in[0];  // (16, 25, 512, 512) f32
  const float* img  = (const float*)d_in[1];  // (16, 1, 512, 512) f32
  float* out        = (float*)d_out;          // (1, 16, 512, 512) f32

  dim3 grid(NH / TILE_H, NB);
  ppconv_kernel<<<grid, NTHREADS, 0, stream>>>(kern, img, out);
}